// FrequencyConv_46926812676470
// MI455X (gfx1250) — compile-verified
//
#include <hip/hip_runtime.h>

// FrequencyConv on MI455X (gfx1250, wave32).
// out[b,o,u,v] = sum_i x[b,i,u,v] * kf[o,i,u,v],
// kf[o,i,u,v] = sum_{p,q<3} (wr + j wi)[o,i,p,q] * exp(-2pi j (u p + v q)/256)
//
// Two-pass, memory-bound design:
//   Pass 1: materialize kf into d_ws (128 MiB, L2-resident: global L2 = 192MB)
//           in WMMA-B-operand order [pixel][lane][16].
//   Pass 2: per-pixel 16x16x16 f32 GEMM via chained v_wmma_f32_16x16x4_f32.
//           x tile in / out tile out via TDM (tensor_load_to_lds /
//           tensor_store_from_lds) with NT cache policy so the streaming
//           operands don't evict kf from L2.
// HBM roofline: x 67MB (NT) + out 134MB (NT) + kf L2-resident ~ 15us @23.3TB/s.

typedef float v2f __attribute__((ext_vector_type(2)));
typedef float v4f __attribute__((ext_vector_type(4)));
typedef float v8f __attribute__((ext_vector_type(8)));
typedef unsigned int v4u __attribute__((ext_vector_type(4)));
typedef int v4i __attribute__((ext_vector_type(4)));
typedef int v8i __attribute__((ext_vector_type(8)));

#define NB   16
#define NCI  16
#define NCO  16
#define HDIM 256
#define WDIM 256
#define PI_F 3.14159265358979323846f

#if __has_builtin(__builtin_amdgcn_tensor_load_to_lds) && \
    __has_builtin(__builtin_amdgcn_tensor_store_from_lds) && \
    __has_builtin(__builtin_amdgcn_s_wait_tensorcnt)
#define USE_TDM 1
#else
#define USE_TDM 0
#endif

#define CPOL_NT 1   // TH[2:0] = NT (evict-first) for streaming operands

// ---------------------------------------------------------------------------
// Pass 1: kf generation. Grid: 2048 blocks = 256 v * 8 u-tiles, 256 threads.
// Block handles fixed v, 32 consecutive u's. Thread = (lane-slot, u-subslot).
// Separable twiddles: wv[o,i,p] = sum_q wc[o,i,p,q]*ev[q]  (once per thread),
// then kf[o,i] = sum_p wv[o,i,p]*eu[p] per u (12 fma per (o,i) pair).
// Output dword order per lane: dw = comp*8 + chunk*2 + reg  (WMMA B layout:
// B reg0 of chunk c holds K=4c (lanes<16) / 4c+2 (lanes>=16), reg1 = +1).
// ---------------------------------------------------------------------------
__global__ __launch_bounds__(256) void kf_gen_kernel(
    const float* __restrict__ w_real,
    const float* __restrict__ w_imag,
    float* __restrict__ kf)
{
    __shared__ float wlds[NCO * NCI * 9 * 2];   // 18 KB, (wr,wi) interleaved

    for (int idx = threadIdx.x; idx < NCO * NCI * 9; idx += 256) {
        wlds[2 * idx + 0] = w_real[idx];
        wlds[2 * idx + 1] = w_imag[idx];
    }
    __syncthreads();

    const int v     = blockIdx.x & 255;
    const int ut    = blockIdx.x >> 8;      // 0..7 -> u base = ut*32
    const int lane  = threadIdx.x & 31;
    const int uslot = threadIdx.x >> 5;     // 0..7
    const int o     = lane & 15;
    const int hi    = lane >> 4;

    // ev[p] = exp(-2pi j v p / 256), p = 0..2
    float evr[3], evi[3];
    {
        float th = -2.0f * PI_F * (float)v * (1.0f / (float)WDIM);
        float s, c;
        __sincosf(th, &s, &c);
        evr[0] = 1.0f; evi[0] = 0.0f;
        evr[1] = c;    evi[1] = s;
        evr[2] = c * c - s * s;
        evi[2] = 2.0f * c * s;
    }

    // wv[chunk][reg][p] : ev folded into this lane's 8 (o,i) weight pairs.
    float wvr[4][2][3], wvi[4][2][3];
#pragma unroll
    for (int c = 0; c < 4; ++c) {
#pragma unroll
        for (int rb = 0; rb < 2; ++rb) {
            const int i = 4 * c + 2 * hi + rb;
            const float* wp = &wlds[(o * NCI + i) * 18];
#pragma unroll
            for (int p = 0; p < 3; ++p) {
                float ar = 0.0f, ai = 0.0f;
#pragma unroll
                for (int q = 0; q < 3; ++q) {
                    const float Wr = wp[(p * 3 + q) * 2 + 0];
                    const float Wi = wp[(p * 3 + q) * 2 + 1];
                    ar += Wr * evr[q] - Wi * evi[q];
                    ai += Wr * evi[q] + Wi * evr[q];
                }
                wvr[c][rb][p] = ar;
                wvi[c][rb][p] = ai;
            }
        }
    }

    // 4 u's per thread
    for (int k = 0; k < 4; ++k) {
        const int u = ut * 32 + k * 8 + uslot;

        float eur[3], eui[3];
        {
            float th = -2.0f * PI_F * (float)u * (1.0f / (float)HDIM);
            float s, c;
            __sincosf(th, &s, &c);
            eur[0] = 1.0f; eui[0] = 0.0f;
            eur[1] = c;    eui[1] = s;
            eur[2] = c * c - s * s;
            eui[2] = 2.0f * c * s;
        }

        float out16[16];
#pragma unroll
        for (int c = 0; c < 4; ++c) {
#pragma unroll
            for (int rb = 0; rb < 2; ++rb) {
                float kr = 0.0f, ki = 0.0f;
#pragma unroll
                for (int p = 0; p < 3; ++p) {
                    kr += wvr[c][rb][p] * eur[p] - wvi[c][rb][p] * eui[p];
                    ki += wvr[c][rb][p] * eui[p] + wvi[c][rb][p] * eur[p];
                }
                out16[0 + c * 2 + rb] = kr;   // comp 0 (real B operand)
                out16[8 + c * 2 + rb] = ki;   // comp 1 (imag B operand)
            }
        }

        const int pixel = u * WDIM + v;
        v4f* dst = (v4f*)(kf + (size_t)(pixel * 32 + lane) * 16);
#pragma unroll
        for (int j = 0; j < 4; ++j) {
            v4f t;
            t[0] = out16[4 * j + 0];
            t[1] = out16[4 * j + 1];
            t[2] = out16[4 * j + 2];
            t[3] = out16[4 * j + 3];
            dst[j] = t;   // 512B coalesced stores per wave; RT -> stays in L2
        }
    }
}

// ---------------------------------------------------------------------------
// TDM descriptor helper (CDNA5 ISA).
// 2D tile: 16 floats per row, tile_rows rows, global row stride = 65536
// floats (one H*W plane per (b,i) / (comp,b,o) image).
// Load path adds 1 dword of LDS padding per 64B row (pad_interval=3,
// pad_amount=0) -> 17-dword LDS row pitch for reduced DS bank conflicts.
// ---------------------------------------------------------------------------
#if USE_TDM
__device__ __forceinline__ void tdm_2d(bool is_store, unsigned lds_byte,
                                       unsigned long long gaddr,
                                       unsigned tile_rows, bool pad)
{
    v4u g0;
    g0[0] = 1u;                                    // count=1, user descriptor
    g0[1] = lds_byte;                              // lds_addr
    g0[2] = (unsigned)(gaddr & 0xffffffffull);     // global_addr[31:0]
    g0[3] = (unsigned)((gaddr >> 32) & 0x1ffffffull) | (2u << 30); // type=2

    v8i g1;
    unsigned w0 = (2u << 16);                      // data_size = 4 bytes
    if (pad) w0 |= (1u << 20) | (3u << 22);        // pad 1 dword per 64B
    g1[0] = (int)w0;                               // wg_mask=0 (not in cluster)
    g1[1] = (int)((65536u & 0xffffu) << 16);       // tensor_dim0 lo16 (=0)
    g1[2] = (int)((65536u >> 16) | (tile_rows << 16)); // td0 hi16 | td1 lo16
    g1[3] = (int)((tile_rows >> 16) | (16u << 16));    // td1 hi16 | tile_dim0
    g1[4] = (int)(tile_rows & 0xffffu);            // tile_dim1 | tile_dim2=0
    g1[5] = (int)65536u;                           // tensor_dim0_stride lo32
    g1[6] = 0;                                     // stride hi | td1_stride lo
    g1[7] = 0;

    v4i g2 = {0, 0, 0, 0};                         // 2D: groups 2/3 unused
    v4i g3 = {0, 0, 0, 0};

#if defined(__clang_major__) && (__clang_major__ >= 23)
    v8i g4 = {0, 0, 0, 0, 0, 0, 0, 0};
    if (is_store)
        __builtin_amdgcn_tensor_store_from_lds(g0, g1, g2, g3, g4, CPOL_NT);
    else
        __builtin_amdgcn_tensor_load_to_lds(g0, g1, g2, g3, g4, CPOL_NT);
#else
    if (is_store)
        __builtin_amdgcn_tensor_store_from_lds(g0, g1, g2, g3, CPOL_NT);
    else
        __builtin_amdgcn_tensor_load_to_lds(g0, g1, g2, g3, CPOL_NT);
#endif
}
#endif

// ---------------------------------------------------------------------------
// Pass 2: per-pixel GEMM. Grid: 4096 blocks = 256 u * 16 v-tiles of 16,
// 256 threads = 8 waves, 2 pixels per wave.
// x tile:  NT TDM load -> xl[bi*17 + v]   (pad trick, 17 KB)
// kf (B):  4 coalesced b128 global loads per pixel, hoisted before the
//          barrier so they fly concurrently with the TDM DMA (L2 hits).
// out:     C -> ol[row*16 + v] -> one NT TDM store of a 16x512 tile.
// ---------------------------------------------------------------------------
__global__ __launch_bounds__(256) void fconv_wmma_kernel(
    const float* __restrict__ x,
    const float* __restrict__ kf,
    float* __restrict__ out)
{
    __shared__ float xl[256 * 17];    // [bi][v + pad], 17 KB
    __shared__ float ol[512 * 16];    // [comp*256 + b*16 + o][v], 32 KB

    const int u  = blockIdx.x >> 4;
    const int v0 = (blockIdx.x & 15) * 16;

    const int lane = threadIdx.x & 31;
    const int wid  = threadIdx.x >> 5;
    const int mb   = lane & 15;     // A row (batch); also C column (cout)
    const int hi   = lane >> 4;

#if USE_TDM
    const bool issuer = (threadIdx.x < 32);
    if (issuer) {                    // one wave issues the DMA (EXEC-ignored op)
        tdm_2d(false, (unsigned)(uintptr_t)(void*)xl,
               (unsigned long long)(uintptr_t)(x + (size_t)u * WDIM + v0),
               256u, true);
    }
#else
    for (int idx = threadIdx.x; idx < 1024; idx += 256) {
        const int bi = idx >> 2;
        const int vq = (idx & 3) * 4;
        const v4f xv = *(const v4f*)(x + (size_t)bi * (HDIM * WDIM)
                                       + u * WDIM + v0 + vq);
#pragma unroll
        for (int j = 0; j < 4; ++j)
            xl[bi * 17 + vq + j] = xv[j];
    }
#endif

    // B fragments for both pixels: issue now, overlap with the x DMA.
    v4f bf[2][4];
#pragma unroll
    for (int pk = 0; pk < 2; ++pk) {
        const int pixel = u * WDIM + v0 + wid * 2 + pk;
        const v4f* bp = (const v4f*)(kf + (size_t)(pixel * 32 + lane) * 16);
#pragma unroll
        for (int j = 0; j < 4; ++j)
            bf[pk][j] = bp[j];
    }

#if USE_TDM
    if (issuer)
        __builtin_amdgcn_s_wait_tensorcnt(0);
#endif
    __syncthreads();

    for (int pk = 0; pk < 2; ++pk) {
        const int vloc = wid * 2 + pk;

        // Unpack B operands (kf already in WMMA-B layout, 64B per lane):
        // dword order comp*8 + chunk*2 + reg.
        v2f br[4], bi_[4];
        br[0][0]  = bf[pk][0][0]; br[0][1]  = bf[pk][0][1];
        br[1][0]  = bf[pk][0][2]; br[1][1]  = bf[pk][0][3];
        br[2][0]  = bf[pk][1][0]; br[2][1]  = bf[pk][1][1];
        br[3][0]  = bf[pk][1][2]; br[3][1]  = bf[pk][1][3];
        bi_[0][0] = bf[pk][2][0]; bi_[0][1] = bf[pk][2][1];
        bi_[1][0] = bf[pk][2][2]; bi_[1][1] = bf[pk][2][3];
        bi_[2][0] = bf[pk][3][0]; bi_[2][1] = bf[pk][3][1];
        bi_[3][0] = bf[pk][3][2]; bi_[3][1] = bf[pk][3][3];

        // A operands from LDS: chunk c -> K rows 4c+2*hi, 4c+2*hi+1.
        v2f a[4];
        const int rowb = mb * 16;
#pragma unroll
        for (int c = 0; c < 4; ++c) {
            const int i0 = 4 * c + 2 * hi;
            a[c][0] = xl[(rowb + i0 + 0) * 17 + vloc];
            a[c][1] = xl[(rowb + i0 + 1) * 17 + vloc];
        }

        v8f cr = {0.f, 0.f, 0.f, 0.f, 0.f, 0.f, 0.f, 0.f};
        v8f ci = {0.f, 0.f, 0.f, 0.f, 0.f, 0.f, 0.f, 0.f};
#pragma unroll
        for (int c = 0; c < 4; ++c) {
            cr = __builtin_amdgcn_wmma_f32_16x16x4_f32(
                     false, a[c], false, br[c], (short)0, cr, false, false);
            ci = __builtin_amdgcn_wmma_f32_16x16x4_f32(
                     false, a[c], false, bi_[c], (short)0, ci, false, false);
        }

        // C fragment: reg j, lanes<16 -> b=j, lanes>=16 -> b=j+8; o = lane&15.
#pragma unroll
        for (int j = 0; j < 8; ++j) {
            const int row0 = (j + 8 * hi) * 16 + mb;
            ol[row0 * 16 + vloc]          = cr[j];
            ol[(256 + row0) * 16 + vloc]  = ci[j];
        }
    }
    __syncthreads();

#if USE_TDM
    if (issuer) {
        tdm_2d(true, (unsigned)(uintptr_t)(void*)ol,
               (unsigned long long)(uintptr_t)(out + (size_t)u * WDIM + v0),
               512u, false);
        __builtin_amdgcn_s_wait_tensorcnt(0);
    }
#else
    for (int idx = threadIdx.x; idx < 512 * 16; idx += 256) {
        const int row = idx >> 4;
        const int vl  = idx & 15;
        out[(size_t)row * (HDIM * WDIM) + u * WDIM + v0 + vl] =
            ol[row * 16 + vl];
    }
#endif
}

extern "C" void kernel_launch(void* const* d_in, const int* in_sizes, int n_in,
                              void* d_out, int out_size, void* d_ws, size_t ws_size,
                              hipStream_t stream) {
    const float* x      = (const float*)d_in[0];
    const float* w_real = (const float*)d_in[1];
    const float* w_imag = (const float*)d_in[2];
    float* out = (float*)d_out;
    float* kf  = (float*)d_ws;   // needs 65536*512*4 = 128 MiB

    kf_gen_kernel<<<2048, 256, 0, stream>>>(w_real, w_imag, kf);
    fconv_wmma_kernel<<<4096, 256, 0, stream>>>(x, kf, out);

    (void)in_sizes; (void)n_in; (void)out_size; (void)ws_size;
}